// Voxelizer_31636729102437
// MI455X (gfx1250) — compile-verified
//
#include <hip/hip_runtime.h>
#include <hip/hip_bf16.h>

typedef __attribute__((ext_vector_type(2))) float v2f;
typedef __attribute__((ext_vector_type(8))) float v8f;

#define GRID_DIM 128       // D = H = W = 128
#define NOFF 125           // 5^3 offsets, MAX_R = 2
#define NTILES 8           // 8 * 16 = 128 offset slots (3 padded)
#define PTS_PER_WAVE 16
#define PTS_PER_BLOCK 128  // 8 waves * 16 points

__global__ void Voxelizer_zero_kernel(float* __restrict__ out, int n) {
    int i = blockIdx.x * blockDim.x + threadIdx.x;
    if (i < n) out[i] = 0.0f;
}

__global__ __launch_bounds__(256) void Voxelizer_31636729102437_kernel(
    const float* __restrict__ pos, const float* __restrict__ scl,
    const float* __restrict__ rot, const float* __restrict__ den,
    float* __restrict__ vol, int N)
{
    // Per wave: 16 points, 16 floats each:
    // [0..11] = W quadratic-form vector (K padded to 12), [12..14] = rounded center, [15] = density
    __shared__ float lds[8][PTS_PER_WAVE][16];

    const int lane = threadIdx.x & 31;
    const int wave = threadIdx.x >> 5;
    const int p0   = blockIdx.x * PTS_PER_BLOCK + wave * PTS_PER_WAVE;

    // ---- Phase 1: lanes 0..15 build per-point W = [q0, 2g, diag(A), 2*offdiag(A)] ----
    if (lane < PTS_PER_WAVE) {
        const int i = p0 + lane;
        float w[12];
        #pragma unroll
        for (int k = 0; k < 12; ++k) w[k] = 0.0f;
        float r0f = 0.f, r1f = 0.f, r2f = 0.f, dn = 0.f;
        if (i < N) {
            const float p0v = pos[i*3+0], p1v = pos[i*3+1], p2v = pos[i*3+2];
            const float s0  = scl[i*3+0], s1  = scl[i*3+1], s2  = scl[i*3+2];
            float qw = rot[i*4+0], qx = rot[i*4+1], qy = rot[i*4+2], qz = rot[i*4+3];
            dn = den[i];

            // voxel-space center; round-to-nearest-even matches jnp.round
            const float c0 = (p0v + 1.0f) * 64.0f - 0.5f;
            const float c1 = (p1v + 1.0f) * 64.0f - 0.5f;
            const float c2 = (p2v + 1.0f) * 64.0f - 0.5f;
            r0f = rintf(c0); r1f = rintf(c1); r2f = rintf(c2);
            const float inv64 = 1.0f / 64.0f;   // shape * 0.5 = 64
            const float f0 = (r0f - c0) * inv64;
            const float f1 = (r1f - c1) * inv64;
            const float f2 = (r2f - c2) * inv64;

            const float qn = 1.0f / sqrtf(qw*qw + qx*qx + qy*qy + qz*qz);
            qw *= qn; qx *= qn; qy *= qn; qz *= qn;
            const float R00 = 1.f - 2.f*(qy*qy + qz*qz), R01 = 2.f*(qx*qy - qw*qz), R02 = 2.f*(qx*qz + qw*qy);
            const float R10 = 2.f*(qx*qy + qw*qz), R11 = 1.f - 2.f*(qx*qx + qz*qz), R12 = 2.f*(qy*qz - qw*qx);
            const float R20 = 2.f*(qx*qz - qw*qy), R21 = 2.f*(qy*qz + qw*qx), R22 = 1.f - 2.f*(qx*qx + qy*qy);

            const float i0 = 1.0f / (s0 + 1e-8f);
            const float i1 = 1.0f / (s1 + 1e-8f);
            const float i2 = 1.0f / (s2 + 1e-8f);
            const float L00 = R00*i0, L01 = R01*i1, L02 = R02*i2;
            const float L10 = R10*i0, L11 = R11*i1, L12 = R12*i2;
            const float L20 = R20*i0, L21 = R21*i1, L22 = R22*i2;

            // A = L L^T (symmetric inverse covariance in normalized coords)
            const float A00 = L00*L00 + L01*L01 + L02*L02;
            const float A01 = L00*L10 + L01*L11 + L02*L12;
            const float A02 = L00*L20 + L01*L21 + L02*L22;
            const float A11 = L10*L10 + L11*L11 + L12*L12;
            const float A12 = L10*L20 + L11*L21 + L12*L22;
            const float A22 = L20*L20 + L21*L21 + L22*L22;

            const float g0 = A00*f0 + A01*f1 + A02*f2;
            const float g1 = A01*f0 + A11*f1 + A12*f2;
            const float g2 = A02*f0 + A12*f1 + A22*f2;
            const float q0 = f0*g0 + f1*g1 + f2*g2;

            w[0] = q0;
            w[1] = 2.f*g0; w[2] = 2.f*g1; w[3] = 2.f*g2;
            w[4] = A00;    w[5] = A11;    w[6] = A22;
            w[7] = 2.f*A01; w[8] = 2.f*A02; w[9] = 2.f*A12;
        }
        #pragma unroll
        for (int k = 0; k < 12; ++k) lds[wave][lane][k] = w[k];
        lds[wave][lane][12] = r0f; lds[wave][lane][13] = r1f;
        lds[wave][lane][14] = r2f; lds[wave][lane][15] = dn;
    }
    __syncthreads();

    // ---- Phase 2: load A fragments per WMMA 16x4 f32 layout ----
    // lanes 0-15: VGPR0=K(kb+0), VGPR1=K(kb+1); lanes 16-31: K(kb+2), K(kb+3)
    const int m16 = lane & 15;
    const int hi  = lane >> 4;   // 0 | 1
    v2f a0, a1, a2;
    a0.x = lds[wave][m16][0 + hi*2]; a0.y = lds[wave][m16][1 + hi*2];
    a1.x = lds[wave][m16][4 + hi*2]; a1.y = lds[wave][m16][5 + hi*2];
    a2.x = lds[wave][m16][8 + hi*2]; a2.y = lds[wave][m16][9 + hi*2];

    // per-lane scatter metadata for the 8 C rows this lane owns (m = r + 8*hi)
    float prz[8], pry[8], prx[8], pdn[8];
    #pragma unroll
    for (int r = 0; r < 8; ++r) {
        const int m = r + 8*hi;
        prz[r] = lds[wave][m][12]; pry[r] = lds[wave][m][13];
        prx[r] = lds[wave][m][14]; pdn[r] = lds[wave][m][15];
    }

    // ---- Phase 3: 8 offset tiles; Mahal(16 pts x 16 offs) = 3 chained f32 WMMAs ----
    #pragma unroll 1
    for (int t = 0; t < NTILES; ++t) {
        const int no = (lane & 15) + t * 16;        // offset column this lane feeds
        const bool offok = (no < NOFF);
        const int ioz = offok ? (no / 25) - 2 : 0;
        const int ioy = offok ? ((no / 5) % 5) - 2 : 0;
        const int iox = offok ? (no % 5) - 2 : 0;
        const float s = 1.0f / 64.0f;
        const float on0 = (float)ioz * s, on1 = (float)ioy * s, on2 = (float)iox * s;

        // constant offset-feature vector u (K layout mirrors A fragments)
        const float u0 = offok ? 1.0f : 0.0f;
        const float u1 = on0, u2 = on1, u3 = on2;
        const float u4 = on0*on0, u5 = on1*on1, u6 = on2*on2;
        const float u7 = on0*on1, u8 = on0*on2, u9 = on1*on2;
        v2f b0, b1, b2;
        b0.x = hi ? u2 : u0;  b0.y = hi ? u3 : u1;
        b1.x = hi ? u6 : u4;  b1.y = hi ? u7 : u5;
        b2.x = hi ? 0.f : u8; b2.y = hi ? 0.f : u9;

        v8f acc = {};
        acc = __builtin_amdgcn_wmma_f32_16x16x4_f32(false, a0, false, b0, (short)0, acc, false, false);
        acc = __builtin_amdgcn_wmma_f32_16x16x4_f32(false, a1, false, b1, (short)0, acc, false, false);
        acc = __builtin_amdgcn_wmma_f32_16x16x4_f32(false, a2, false, b2, (short)0, acc, false, false);

        // ---- scatter: C reg r, this lane -> point m = r + 8*hi, offset column no ----
        if (offok) {
            #pragma unroll
            for (int r = 0; r < 8; ++r) {
                const float mah = acc[r];
                const int gi = p0 + r + 8*hi;
                if (gi < N && mah <= 9.0f) {
                    const int gz = (int)prz[r] + ioz;
                    const int gy = (int)pry[r] + ioy;
                    const int gx = (int)prx[r] + iox;
                    if ((unsigned)gz < (unsigned)GRID_DIM &&
                        (unsigned)gy < (unsigned)GRID_DIM &&
                        (unsigned)gx < (unsigned)GRID_DIM) {
                        const float val = __expf(-0.5f * mah) * pdn[r];
                        atomicAdd(vol + (gz << 14) + (gy << 7) + gx, val);
                    }
                }
            }
        }
    }
}

extern "C" void kernel_launch(void* const* d_in, const int* in_sizes, int n_in,
                              void* d_out, int out_size, void* d_ws, size_t ws_size,
                              hipStream_t stream) {
    const float* pos = (const float*)d_in[0];
    const float* scl = (const float*)d_in[1];
    const float* rot = (const float*)d_in[2];
    const float* den = (const float*)d_in[3];
    const int N = in_sizes[3];          // density count = N_PTS
    float* vol = (float*)d_out;

    // output is poisoned / accumulated into; zero it every call (graph-replay safe)
    Voxelizer_zero_kernel<<<(out_size + 255) / 256, 256, 0, stream>>>(vol, out_size);

    const int blocks = (N + PTS_PER_BLOCK - 1) / PTS_PER_BLOCK;
    Voxelizer_31636729102437_kernel<<<blocks, 256, 0, stream>>>(pos, scl, rot, den, vol, N);
}